// EGCNConv_53755810676780
// MI455X (gfx1250) — compile-verified
//
#include <hip/hip_runtime.h>
#include <hip/hip_bf16.h>

typedef __attribute__((ext_vector_type(2))) float v2f;
typedef __attribute__((ext_vector_type(8))) float v8f;

#define WAVES_PER_BLOCK 8
#define BLOCK_THREADS   (WAVES_PER_BLOCK * 32)

// D = A(16x4) x B(4x16) + C, f32, per CDNA5 ISA layout.
// A: lane m=L&15 -> row M=m ; hi=L>>4 selects K+2 ; vgpr0=K, vgpr1=K+1
// B: lane m -> col N=m ; hi selects K+2                (B[k][n] = W[n][k])
// C/D: vgpr i, lane L -> (M = i + 8*hi, N = L&15)
static __device__ __forceinline__ v8f wmma_f32_k4(v2f a, v2f b, v8f c) {
    return __builtin_amdgcn_wmma_f32_16x16x4_f32(
        /*neg_a=*/false, a, /*neg_b=*/false, b,
        /*c_mod=*/(short)0, c, /*reuse_a=*/false, /*reuse_b=*/false);
}

// ---------------------------------------------------------------------------
// Kernel 0: zero h_out accumulator (N*64 f32 in d_out) + degree (N f32 in ws)
// ---------------------------------------------------------------------------
__global__ void egcn_zero(float* __restrict__ out_h, float* __restrict__ deg, int N) {
    int gid = blockIdx.x * blockDim.x + threadIdx.x;
    int total4 = N * 16;                       // N*64 floats as float4
    if (gid < total4) {
        float4 z = {0.f, 0.f, 0.f, 0.f};
        ((float4*)out_h)[gid] = z;
    }
    if (gid < N) deg[gid] = 0.f;
}

// ---------------------------------------------------------------------------
// Kernel 1: node transforms. Y = X @ W.T for W_ni, W_nj, W_node (+b_node).
// One wave per 16-node tile. Weights staged in LDS.
// ---------------------------------------------------------------------------
__global__ void egcn_node_transform(const float* __restrict__ X,
                                    const float* __restrict__ Wni,
                                    const float* __restrict__ Wnj,
                                    const float* __restrict__ Wnode,
                                    const float* __restrict__ bnode,
                                    float* __restrict__ f_ni,
                                    float* __restrict__ f_nj,
                                    float* __restrict__ h,
                                    int N) {
    __shared__ float sW[3 * 64 * 64];
    __shared__ float sB[64];
    const int tid = threadIdx.x;
    for (int i = tid; i < 4096; i += BLOCK_THREADS) {
        sW[i]        = Wni[i];
        sW[4096 + i] = Wnj[i];
        sW[8192 + i] = Wnode[i];
    }
    if (tid < 64) sB[tid] = bnode[tid];
    __syncthreads();

    const int wave = tid >> 5;
    const int lane = tid & 31;
    const int m    = lane & 15;
    const int hi   = lane >> 4;

    const int tile = blockIdx.x * WAVES_PER_BLOCK + wave;
    const int rbase = tile * 16;
    const int row  = rbase + m;
    const int rclamp = row < N ? row : (N - 1);
    const bool full = (rbase + 16) <= N;       // wave-uniform

    // A fragments: 16 k-blocks covering K=64
    v2f a[16];
    const float* xr = X + (size_t)rclamp * 64 + 2 * hi;
#pragma unroll
    for (int k = 0; k < 16; ++k) a[k] = *(const v2f*)(xr + 4 * k);

    float* const outs[3] = {f_ni, f_nj, h};
#pragma unroll
    for (int w = 0; w < 3; ++w) {
        const float* wbase0 = sW + w * 4096;
        float* const O = outs[w];
#pragma unroll
        for (int nb = 0; nb < 4; ++nb) {
            v8f acc = {};
            const float* wb = wbase0 + (nb * 16 + m) * 64 + 2 * hi;
#pragma unroll
            for (int k = 0; k < 16; ++k) {
                v2f b = *(const v2f*)(wb + 4 * k);   // ds_load_b64
                acc = wmma_f32_k4(a[k], b, acc);
            }
            const int col = nb * 16 + m;
            if (w == 2) {                            // bias only on W_node pass
                const float badd = sB[col];
#pragma unroll
                for (int i = 0; i < 8; ++i) acc[i] += badd;
            }
            float* const obase = O + (size_t)(rbase + 8 * hi) * 64 + col;
            if (full) {                              // scalar branch, no exec churn
#pragma unroll
                for (int i = 0; i < 8; ++i) obase[(size_t)i * 64] = acc[i];
            } else {
#pragma unroll
                for (int i = 0; i < 8; ++i) {
                    if (rbase + 8 * hi + i < N) obase[(size_t)i * 64] = acc[i];
                }
            }
        }
    }
}

// ---------------------------------------------------------------------------
// Kernel 2: edge update. f_out = leaky_relu(E @ Wfij.T + f_ni[src] + f_nj[dst] + bias)
// One wave per 16-edge tile.
// ---------------------------------------------------------------------------
__global__ void egcn_edge_transform(const float* __restrict__ Ef,
                                    const int* __restrict__ src,
                                    const int* __restrict__ dst,
                                    const float* __restrict__ Wfij,
                                    const float* __restrict__ bias,
                                    const float* __restrict__ f_ni,
                                    const float* __restrict__ f_nj,
                                    float* __restrict__ f_out,
                                    int E) {
    __shared__ float sW[64 * 64];
    __shared__ float sB[64];
    const int tid = threadIdx.x;
    for (int i = tid; i < 4096; i += BLOCK_THREADS) sW[i] = Wfij[i];
    if (tid < 64) sB[tid] = bias[tid];
    __syncthreads();

    const int wave = tid >> 5;
    const int lane = tid & 31;
    const int m    = lane & 15;
    const int hi   = lane >> 4;

    const int tile  = blockIdx.x * WAVES_PER_BLOCK + wave;
    const int ebase = tile * 16;
    const int row   = ebase + m;
    const int rclamp = row < E ? row : (E - 1);
    const bool full = (ebase + 16) <= E;       // wave-uniform

    v2f a[16];
    const float* er = Ef + (size_t)rclamp * 64 + 2 * hi;
#pragma unroll
    for (int k = 0; k < 16; ++k) a[k] = *(const v2f*)(er + 4 * k);

    // indices for the 8 edge rows this lane contributes to (C/D layout rows)
    int si[8], di[8];
#pragma unroll
    for (int i = 0; i < 8; ++i) {
        int e  = ebase + i + 8 * hi;
        int ec = e < E ? e : (E - 1);
        si[i] = src[ec];
        di[i] = dst[ec];
    }

#pragma unroll
    for (int nb = 0; nb < 4; ++nb) {
        v8f acc = {};
        const float* wb = sW + (nb * 16 + m) * 64 + 2 * hi;
#pragma unroll
        for (int k = 0; k < 16; ++k) {
            v2f b = *(const v2f*)(wb + 4 * k);
            acc = wmma_f32_k4(a[k], b, acc);
        }
        const int col = nb * 16 + m;
        const float bv = sB[col];
        float* const obase = f_out + (size_t)(ebase + 8 * hi) * 64 + col;
        if (full) {
#pragma unroll
            for (int i = 0; i < 8; ++i) {
                float v = acc[i] + bv
                        + f_ni[(size_t)si[i] * 64 + col]
                        + f_nj[(size_t)di[i] * 64 + col];
                v = v > 0.f ? v : 0.01f * v;     // leaky_relu(0.01)
                obase[(size_t)i * 64] = v;
            }
        } else {
#pragma unroll
            for (int i = 0; i < 8; ++i) {
                if (ebase + 8 * hi + i < E) {
                    float v = acc[i] + bv
                            + f_ni[(size_t)si[i] * 64 + col]
                            + f_nj[(size_t)di[i] * 64 + col];
                    v = v > 0.f ? v : 0.01f * v;
                    obase[(size_t)i * 64] = v;
                }
            }
        }
    }
}

// ---------------------------------------------------------------------------
// Kernel 3: scatter-add h[src] into out_h[dst] + degree. Thread = (edge, 4-col group)
// ---------------------------------------------------------------------------
__global__ void egcn_scatter(const float* __restrict__ h,
                             const int* __restrict__ src,
                             const int* __restrict__ dst,
                             float* __restrict__ out_h,
                             float* __restrict__ deg,
                             int E) {
    long gid = (long)blockIdx.x * blockDim.x + threadIdx.x;
    if (gid >= (long)E * 16) return;
    int e  = (int)(gid >> 4);
    int cg = (int)(gid & 15);
    int s = src[e];
    int d = dst[e];
    const float4 hv = *(const float4*)(h + (size_t)s * 64 + cg * 4);
    float* o = out_h + (size_t)d * 64 + cg * 4;
    unsafeAtomicAdd(o + 0, hv.x);
    unsafeAtomicAdd(o + 1, hv.y);
    unsafeAtomicAdd(o + 2, hv.z);
    unsafeAtomicAdd(o + 3, hv.w);
    if (cg == 0) unsafeAtomicAdd(deg + d, 1.0f);
}

// ---------------------------------------------------------------------------
// Kernel 4: h_out = msg_sum / max(deg, 1)
// ---------------------------------------------------------------------------
__global__ void egcn_finalize(float* __restrict__ out_h,
                              const float* __restrict__ deg, int N) {
    int gid = blockIdx.x * blockDim.x + threadIdx.x;
    if (gid < N * 64) {
        int n = gid >> 6;
        out_h[gid] = out_h[gid] / fmaxf(deg[n], 1.0f);
    }
}

// ---------------------------------------------------------------------------
extern "C" void kernel_launch(void* const* d_in, const int* in_sizes, int n_in,
                              void* d_out, int out_size, void* d_ws, size_t ws_size,
                              hipStream_t stream) {
    const float* nfeats = (const float*)d_in[0];
    const float* efeats = (const float*)d_in[1];
    const int*   src    = (const int*)d_in[2];
    const int*   dst    = (const int*)d_in[3];
    const float* W_node = (const float*)d_in[4];
    const float* b_node = (const float*)d_in[5];
    const float* W_ni   = (const float*)d_in[6];
    const float* W_nj   = (const float*)d_in[7];
    const float* W_fij  = (const float*)d_in[8];
    const float* bias   = (const float*)d_in[9];

    const int N = in_sizes[0] / 64;
    const int E = in_sizes[2];

    float* out_h = (float*)d_out;                       // [N,64]
    float* f_out = (float*)d_out + (size_t)N * 64;      // [E,64]

    float* ws   = (float*)d_ws;
    float* f_ni = ws;                                   // [N,64]
    float* f_nj = ws + (size_t)N * 64;                  // [N,64]
    float* h    = ws + (size_t)2 * N * 64;              // [N,64]
    float* deg  = ws + (size_t)3 * N * 64;              // [N]

    // 0: zero accumulators
    {
        int total = N * 16;                             // float4 count (>= N)
        int blocks = (total + 255) / 256;
        egcn_zero<<<blocks, 256, 0, stream>>>(out_h, deg, N);
    }
    // 1: node transforms (f_ni, f_nj, h)
    {
        int tiles  = (N + 15) / 16;
        int blocks = (tiles + WAVES_PER_BLOCK - 1) / WAVES_PER_BLOCK;
        egcn_node_transform<<<blocks, BLOCK_THREADS, 0, stream>>>(
            nfeats, W_ni, W_nj, W_node, b_node, f_ni, f_nj, h, N);
    }
    // 2: edge transform + gather + leaky_relu
    {
        int tiles  = (E + 15) / 16;
        int blocks = (tiles + WAVES_PER_BLOCK - 1) / WAVES_PER_BLOCK;
        egcn_edge_transform<<<blocks, BLOCK_THREADS, 0, stream>>>(
            efeats, src, dst, W_fij, bias, f_ni, f_nj, f_out, E);
    }
    // 3: scatter h[src] -> out_h[dst], degree
    {
        long total = (long)E * 16;
        int blocks = (int)((total + 255) / 256);
        egcn_scatter<<<blocks, 256, 0, stream>>>(h, src, dst, out_h, deg, E);
    }
    // 4: mean
    {
        int blocks = (N * 64 + 255) / 256;
        egcn_finalize<<<blocks, 256, 0, stream>>>(out_h, deg, N);
    }
}